// video_denoise_61375082660121
// MI455X (gfx1250) — compile-verified
//
#include <hip/hip_runtime.h>

// CDNA5 (gfx1250) video-denoise: two chained blocks of
//   tanh(a*conv5x5_d1 + b*conv5x5_d2 + c*conv5x5_d4) over (prev,cur,next) frames,
// reflect padding. f32 conv mapped onto V_WMMA_F32_16X16X4_F32 via a band
// (Toeplitz) matrix formulation: out[m,c] += sum_j A[m,j] * B[j,c],
//   A[m,j] = x[ch][m + dil*(kr-2)][j-8]   (16 x 32 band, in K=4 chunks)
//   B[j,c] = scale*w[ch,kr,kc]  iff  j == c + 8 + dil*(kc-2)
// The per-lane B pair depends only on (conv, dl0 = 4q+klocal-8-n), so all B
// values are pre-packed into a small LDS table (plus one zero window); the hot
// loop is pure  ds_load_b64(A) + ds_load_b64(B) + v_wmma, no exec masking.

typedef float v2f __attribute__((ext_vector_type(2)));
typedef float v8f __attribute__((ext_vector_type(8)));

namespace {
constexpr int Bn = 4, Tn = 16, Hn = 512, Wn = 512;
constexpr int TILE_H = 32;                  // rows per workgroup
constexpr int TILE_W = 64;                  // cols per workgroup
constexpr int HALO   = 8;                   // max reach: dil=4, 5 taps -> +/-8
constexpr int LH     = TILE_H + 2 * HALO;   // 48
constexpr int LW     = TILE_W + 2 * HALO;   // 80
constexpr int LSTR   = LW + 2;              // 82 (even -> 8B-aligned b64 LDS loads)
constexpr int CHS    = LH * LSTR;           // channel stride in s_in (floats)

// Packed-B windows: [zero][conv0: dl0 in -3..2][conv1: -5..4][conv2: -9..8]
constexpr int W0LO = -3, W0N = 6;
constexpr int W1LO = -5, W1N = 10;
constexpr int W2LO = -9, W2N = 18;
constexpr int WB0 = 1;                      // window index bases (0 = zero window)
constexpr int WB1 = WB0 + W0N;              // 7
constexpr int WB2 = WB1 + W1N;              // 17
constexpr int NWIN = WB2 + W2N;             // 35
constexpr int WSTR = 30;                    // floats per window: 15 (ch,kr) pairs
}

__device__ __forceinline__ int reflect_idx(int i, int n) {
    if (i < 0)  i = -i;
    if (i >= n) i = 2 * n - 2 - i;
    return i;
}

__global__ __launch_bounds__(256)
void denoise_block_wmma(const float* __restrict__ x, float* __restrict__ y,
                        const float* __restrict__ w1, const float* __restrict__ b1,
                        const float* __restrict__ w2, const float* __restrict__ b2,
                        const float* __restrict__ w3, const float* __restrict__ b3,
                        const float* __restrict__ sa, const float* __restrict__ sb,
                        const float* __restrict__ sc)
{
    __shared__ float s_in[3 * CHS];          // 3 halo tiles, 46.1 KB
    __shared__ float s_bt[NWIN * WSTR];      // packed B table, 4.2 KB

    const int tid = threadIdx.x;
    const int b   = blockIdx.z;
    const int t   = blockIdx.y;
    const int tx  = blockIdx.x & 7;          // 512/64 = 8 tiles wide
    const int ty  = blockIdx.x >> 3;         // 512/32 = 16 tiles tall
    const int r0  = ty * TILE_H;
    const int c0  = tx * TILE_W;

    const float sc0 = sa[0], sc1 = sb[0], sc2 = sc[0];
    const float biasT = sc0 * b1[0] + sc1 * b2[0] + sc2 * b3[0];

    // ---- Build packed B table: value = scale * w[ch][kr][kc] (masked) ----
    for (int i = tid; i < NWIN * WSTR; i += 256) {
        const int widx  = i / WSTR;
        const int r     = i - widx * WSTR;
        const int pair  = r >> 1;            // 0..14 = ch*5 + kr
        const int which = r & 1;             // bv0 / bv1
        float val = 0.f;
        if (widx != 0) {
            int conv, dl0;
            if (widx < WB1)      { conv = 0; dl0 = (widx - WB0) + W0LO; }
            else if (widx < WB2) { conv = 1; dl0 = (widx - WB1) + W1LO; }
            else                 { conv = 2; dl0 = (widx - WB2) + W2LO; }
            const int dil   = 1 << conv;
            const int delta = dl0 + which;
            const int kc    = (delta >> conv) + 2;   // exact when divisible
            const bool ok   = ((delta & (dil - 1)) == 0) && (kc >= 0) && (kc <= 4);
            if (ok) {
                const float* wp = (conv == 0) ? w1 : (conv == 1) ? w2 : w3;
                const float  s  = (conv == 0) ? sc0 : (conv == 1) ? sc1 : sc2;
                val = s * wp[pair * 5 + kc];         // pair*5 = ch*25 + kr*5
            }
        }
        s_bt[i] = val;
    }

    // ---- Stage triplet frames' halo tiles into LDS (reflect pad) ----
    const int tprev = (t == 0) ? 1 : t - 1;
    const int tnext = (t == Tn - 1) ? Tn - 2 : t + 1;
    const float* f0 = x + (size_t)(b * Tn + tprev) * Hn * Wn;
    const float* f1 = x + (size_t)(b * Tn + t    ) * Hn * Wn;
    const float* f2 = x + (size_t)(b * Tn + tnext) * Hn * Wn;

    for (int i = tid; i < 3 * LH * LW; i += 256) {
        const int ch  = i / (LH * LW);
        const int rem = i - ch * (LH * LW);
        const int rr  = rem / LW;
        const int cc  = rem - rr * LW;
        const int gr  = reflect_idx(r0 - HALO + rr, Hn);
        const int gc  = reflect_idx(c0 - HALO + cc, Wn);
        const float* src = (ch == 0) ? f0 : (ch == 1) ? f1 : f2;
        s_in[ch * CHS + rr * LSTR + cc] = src[(size_t)gr * Wn + gc];
    }
    __syncthreads();

    // ---- Per-wave 16x16 output sub-tile via f32 WMMA ----
    const int lane   = tid & 31;
    const int wave   = tid >> 5;
    const int wr     = wave >> 2;            // 0..1 (row sub-tile)
    const int wc     = wave & 3;             // 0..3 (col sub-tile)
    const int nIdx   = lane & 15;            // N for B/D, M for A
    const int half   = lane >> 4;
    const int klocal = half * 2;             // this lane's K base within a chunk

    // A base biased to drow = -8 so all ds offsets are non-negative immediates:
    // element for (ch, kr, q, v) = aLane[4*q + ch*CHS + (dil*(kr-2)+8)*LSTR + v]
    const float* aLane = &s_in[(wr * 16 + nIdx) * LSTR + wc * 16 + klocal];
    const int dlLane = klocal - 8 - nIdx;    // dl0 minus 4*q

    v8f acc = {0.f, 0.f, 0.f, 0.f, 0.f, 0.f, 0.f, 0.f};

    #pragma unroll
    for (int conv = 0; conv < 3; ++conv) {
        const int dil   = 1 << conv;
        const int qlo   = (conv == 2) ? 0 : 1;      // skip all-zero B chunks
        const int qhi   = (conv == 2) ? 7 : 6;
        const int wlo   = (conv == 0) ? W0LO : (conv == 1) ? W1LO : W2LO;
        const int wn    = (conv == 0) ? W0N  : (conv == 1) ? W1N  : W2N;
        const int wbase = (conv == 0) ? WB0  : (conv == 1) ? WB1  : WB2;

        #pragma unroll
        for (int q = qlo; q <= qhi; ++q) {
            const int dl0 = 4 * q + dlLane;
            const int sel = dl0 - wlo;
            const bool inw = (unsigned)sel < (unsigned)wn;
            const int woff = inw ? (wbase + sel) * WSTR : 0;   // zero window else
            const v2f* bb = (const v2f*)&s_bt[woff];           // 8B aligned
            const float* ab = aLane + 4 * q;

            #pragma unroll
            for (int ch = 0; ch < 3; ++ch) {
                #pragma unroll
                for (int kr = 0; kr < 5; ++kr) {
                    const v2f av = *(const v2f*)&ab[ch * CHS + (dil * (kr - 2) + 8) * LSTR];
                    const v2f bv = bb[ch * 5 + kr];
                    acc = __builtin_amdgcn_wmma_f32_16x16x4_f32(
                        false, av, false, bv, (short)0, acc, false, false);
                }
            }
        }
    }

    // ---- bias + tanh + store (D layout: m = v + 8*half, n = nIdx) ----
    float* outf = y + (size_t)(b * Tn + t) * Hn * Wn;
    #pragma unroll
    for (int v = 0; v < 8; ++v) {
        const int m = v + 8 * half;
        const float val = acc[v] + biasT;
        const float e = __expf(2.f * val);           // tanh = 1 - 2/(e^{2x}+1)
        const float th = 1.f - 2.f / (e + 1.f);      // correct limits at +/-inf
        outf[(size_t)(r0 + wr * 16 + m) * Wn + (c0 + wc * 16 + nIdx)] = th;
    }
}

extern "C" void kernel_launch(void* const* d_in, const int* in_sizes, int n_in,
                              void* d_out, int out_size, void* d_ws, size_t ws_size,
                              hipStream_t stream)
{
    // setup_inputs order:
    // 0:x, 1..6: blk1 w1,b1,w2,b2,w3,b3, 7..9: blk1 a,b,c,
    // 10..15: blk2 w1,b1,w2,b2,w3,b3, 16..18: blk2 a,b,c
    const float* p[19];
    for (int i = 0; i < 19 && i < n_in; ++i) p[i] = (const float*)d_in[i];

    float* y = (float*)d_ws;        // intermediate block-1 output, 64 MB
    float* z = (float*)d_out;

    dim3 grid(128, Tn, Bn);         // 8x16 tiles per 512x512 frame
    dim3 block(256);                // 8 waves -> 2x4 grid of 16x16 WMMA sub-tiles

    denoise_block_wmma<<<grid, block, 0, stream>>>(
        p[0], y, p[1], p[2], p[3], p[4], p[5], p[6], p[7], p[8], p[9]);
    denoise_block_wmma<<<grid, block, 0, stream>>>(
        y, z, p[10], p[11], p[12], p[13], p[14], p[15], p[16], p[17], p[18]);
}